// MinibatchLayer_4020089389082
// MI455X (gfx1250) — compile-verified
//
#include <hip/hip_runtime.h>
#include <hip/hip_bf16.h>

// Problem constants (from setup_inputs): B=512, NIN=1024, NK=128, DK=5
#define BSZ  512
#define NIN  1024
#define NK   128
#define DK   5
#define KD   (NK * DK)      // 640 flattened output columns of the GEMM
#define NOUT (NIN + NK)     // 1152

typedef float v2f __attribute__((ext_vector_type(2)));
typedef float v8f __attribute__((ext_vector_type(8)));

// ---------------------------------------------------------------------------
// Kernel 1: scale[c] = exp(lws[c]) / sqrt(sum_i theta[i, c]^2),  c in [0,640)
// One wave32 per column; column stride through theta is KD floats.
// ---------------------------------------------------------------------------
__global__ void mb_colnorm(const float* __restrict__ theta,
                           const float* __restrict__ lws,
                           float* __restrict__ scale) {
    const int wave = (blockIdx.x * blockDim.x + threadIdx.x) >> 5;
    const int lane = threadIdx.x & 31;
    if (wave >= KD) return;
    float s = 0.0f;
    #pragma unroll 4
    for (int i = lane; i < NIN; i += 32) {
        float t = theta[i * KD + wave];
        s += t * t;
    }
    #pragma unroll
    for (int off = 16; off > 0; off >>= 1)
        s += __shfl_xor(s, off, 32);
    if (lane == 0)
        scale[wave] = expf(lws[wave]) / sqrtf(s);
}

// ---------------------------------------------------------------------------
// Kernel 2: actv[b, c] = scale[c] * sum_i x[b,i] * theta[i,c]
// f32 WMMA 16x16x4: one wave per 16x16 C tile, K swept 4 at a time.
//   A (16x4):  lanes 0-15 hold {K=0,K=1}, lanes 16-31 hold {K=2,K=3} (b64 load)
//   B (4x16):  lane n holds column N=n; v0/v1 = K pair per half-wave
//   C (16x16): VGPR j, lanes 0-15 -> M=j, lanes 16-31 -> M=j+8
// ---------------------------------------------------------------------------
__global__ void mb_gemm_wmma(const float* __restrict__ x,      // [512,1024]
                             const float* __restrict__ theta,  // [1024,640]
                             const float* __restrict__ scale,  // [640]
                             float* __restrict__ actv) {       // [512,640]
    const int lane    = threadIdx.x & 31;
    const int tile    = blockIdx.x * (blockDim.x >> 5) + (threadIdx.x >> 5);
    const int tiles_n = KD / 16;              // 40
    const int tm      = tile / tiles_n;       // 0..31
    const int tn      = tile % tiles_n;       // 0..39

    const int l15   = lane & 15;
    const int khalf = (lane >> 4) << 1;       // 0 or 2

    const int row = tm * 16 + l15;            // A row this lane feeds
    const int col = tn * 16 + l15;            // B/C column this lane feeds

    const float* aptr = x + row * NIN + khalf;
    const float* bptr = theta + khalf * KD + col;

    v8f acc = {};
    #pragma unroll 8
    for (int k = 0; k < NIN; k += 4) {
        v2f a = *(const v2f*)aptr;            // x[row, k+khalf .. k+khalf+1]
        v2f b;
        b.x = bptr[0];                        // theta[k+khalf,   col]
        b.y = bptr[KD];                       // theta[k+khalf+1, col]
        acc = __builtin_amdgcn_wmma_f32_16x16x4_f32(
            /*neg_a=*/false, a, /*neg_b=*/false, b,
            /*c_mod=*/(short)0, acc, /*reuse_a=*/false, /*reuse_b=*/false);
        aptr += 4;
        bptr += 4 * KD;
    }

    const float s     = scale[col];
    const int   mbase = tm * 16 + ((lane >> 4) << 3);
    #pragma unroll
    for (int j = 0; j < 8; ++j)
        actv[(mbase + j) * KD + col] = acc[j] * s;
}

// ---------------------------------------------------------------------------
// Kernel 3: f[b,k] = sum_{b2} exp(-L1(actv[b,k,:], actv[b2,k,:])) - 1 + bias[k]
// (diagonal: reference adds 1e6 penalty -> exp underflows to 0; our loop
//  includes b2==b contributing exp(0)=1, so subtract exactly 1.0)
// One block per k (128 blocks), 512 threads = one per b; slice staged in LDS.
// ---------------------------------------------------------------------------
__global__ void mb_pairwise(const float* __restrict__ actv,  // [512,640]
                            const float* __restrict__ bias,  // [128]
                            float* __restrict__ out) {       // [512,1152]
    const int k = blockIdx.x;
    const int b = threadIdx.x;
    __shared__ float s[BSZ * DK];             // 10 KB

    const float* ap = actv + b * KD + k * DK;
    const float a0 = ap[0], a1 = ap[1], a2 = ap[2], a3 = ap[3], a4 = ap[4];
    s[b * DK + 0] = a0; s[b * DK + 1] = a1; s[b * DK + 2] = a2;
    s[b * DK + 3] = a3; s[b * DK + 4] = a4;
    __syncthreads();

    float f = 0.0f;
    #pragma unroll 4
    for (int b2 = 0; b2 < BSZ; ++b2) {
        const float* q = s + b2 * DK;
        float d = fabsf(a0 - q[0]) + fabsf(a1 - q[1]) + fabsf(a2 - q[2]) +
                  fabsf(a3 - q[3]) + fabsf(a4 - q[4]);
        f += __expf(-d);
    }
    out[b * NOUT + NIN + k] = (f - 1.0f) + bias[k];
}

// ---------------------------------------------------------------------------
// Kernel 4: out[b, 0:1024] = x[b, :]   (float4-vectorized copy)
// ---------------------------------------------------------------------------
__global__ void mb_copyx(const float* __restrict__ x, float* __restrict__ out) {
    const int idx = blockIdx.x * blockDim.x + threadIdx.x;   // over 512*256 f4
    const int row = idx >> 8;                 // / (NIN/4)
    const int c4  = idx & 255;
    reinterpret_cast<float4*>(out + row * NOUT)[c4] =
        reinterpret_cast<const float4*>(x + row * NIN)[c4];
}

extern "C" void kernel_launch(void* const* d_in, const int* in_sizes, int n_in,
                              void* d_out, int out_size, void* d_ws, size_t ws_size,
                              hipStream_t stream) {
    const float* x     = (const float*)d_in[0];   // [512,1024]
    const float* theta = (const float*)d_in[1];   // [1024,128,5] == [1024,640]
    const float* lws   = (const float*)d_in[2];   // [640]
    const float* bias  = (const float*)d_in[3];   // [128]
    float* out = (float*)d_out;                   // [512,1152]

    float* scale = (float*)d_ws;                  // 640 floats
    float* actv  = scale + 1024;                  // [512,640] floats (aligned)

    // 640 columns, one wave each: 80 blocks x 256 threads (8 waves)
    mb_colnorm<<<80, 256, 0, stream>>>(theta, lws, scale);

    // 32x40 = 1280 16x16 tiles, one wave each: 320 blocks x 128 threads
    mb_gemm_wmma<<<320, 128, 0, stream>>>(x, theta, scale, actv);

    // one block per k, one thread per b
    mb_pairwise<<<NK, BSZ, 0, stream>>>(actv, bias, out);

    // 512*1024 floats as float4: 512 blocks x 256 threads
    mb_copyx<<<512, 256, 0, stream>>>(x, out);
}